// OverlapLoss_75608604279391
// MI455X (gfx1250) — compile-verified
//
#include <hip/hip_runtime.h>
#include <stdint.h>

// Problem constants (from the reference): groups of 22, top-11, scalar out.
#define GROUP 22
#define TOPK  11
#define GPB   256                 // groups per block (one group per thread)
#define TILE  (GPB * GROUP)       // 5632 floats per array per block tile
#define TILE_V4 (TILE / 4)        // 1408 16-byte vectors per array

typedef __attribute__((ext_vector_type(2))) float v2f;
typedef __attribute__((ext_vector_type(8))) float v8f;

// ---------------------------------------------------------------------------
// Kernel 1: stream tiles into LDS via CDNA5 async-LDS loads, then each thread
// ranks its 22-element group in registers (exact lax.top_k tie semantics) and
// accumulates (mse_partial, overlap_partial). Deterministic block reduction.
// ---------------------------------------------------------------------------
__global__ __launch_bounds__(256)
void overlap_main(const float* __restrict__ pred,
                  const float* __restrict__ targ,
                  float* __restrict__ partials,
                  int M, long long Nll) {
  __shared__ __align__(16) float sp[TILE];
  __shared__ __align__(16) float st[TILE];
  __shared__ float red[16];

  const int tid = threadIdx.x;
  const long long tileBase = (long long)blockIdx.x * TILE;  // element index
  const long long Nv4 = Nll >> 2;                           // full vec4 count
  // Generic shared-aperture pointers have the LDS byte offset in their low 32
  // bits (ISA 10.2).
  const unsigned lds_p = (unsigned)(size_t)(void*)sp;
  const unsigned lds_t = (unsigned)(size_t)(void*)st;

  // ---- Async copy global -> LDS (ASYNCcnt-tracked, bypasses VGPRs) ----
  for (int v = tid; v < TILE_V4; v += 256) {
    long long g4 = (tileBase >> 2) + v;
    if (g4 < Nv4) {
      unsigned dp = lds_p + ((unsigned)v << 4);
      unsigned dt = lds_t + ((unsigned)v << 4);
      uint64_t ap = (uint64_t)(const void*)(pred + (g4 << 2));
      uint64_t at = (uint64_t)(const void*)(targ + (g4 << 2));
      asm volatile("global_load_async_to_lds_b128 %0, %1, off"
                   :: "v"(dp), "v"(ap) : "memory");
      asm volatile("global_load_async_to_lds_b128 %0, %1, off"
                   :: "v"(dt), "v"(at) : "memory");
    }
  }
  // Scalar tail (only taken if N % 4 != 0; N = 11,000,000 is divisible by 4).
  {
    long long tailStart = Nv4 << 2;
    for (long long e = tileBase + tid; e < tileBase + TILE; e += 256) {
      if (e >= tailStart && e < Nll) {
        sp[e - tileBase] = pred[e];
        st[e - tileBase] = targ[e];
      }
    }
  }
  asm volatile("s_wait_asynccnt 0x0" ::: "memory");
  __syncthreads();

  // ---- Per-thread group processing entirely in registers ----
  float mse = 0.0f;
  int ovl = 0;
  const int g = blockIdx.x * GPB + tid;
  if (g < M) {
    float p[GROUP], t[GROUP];
    const float* lp = sp + tid * GROUP;
    const float* lt = st + tid * GROUP;
#pragma unroll
    for (int i = 0; i < GROUP; ++i) {
      p[i] = lp[i];
      t[i] = lt[i];
      float d = p[i] - t[i];
      mse += d * d;
    }
    int rp[GROUP], rt[GROUP];
#pragma unroll
    for (int i = 0; i < GROUP; ++i) { rp[i] = 0; rt[i] = 0; }
    // rank_i under (value desc, index asc): for pair i<j,
    //  p[i] >= p[j]  -> j is outranked by i (strictly greater, or tie with
    //                   smaller index) -> rp[j]++ ; else rp[i]++.
#pragma unroll
    for (int i = 0; i < GROUP; ++i) {
#pragma unroll
      for (int j = i + 1; j < GROUP; ++j) {
        int cp = (p[i] >= p[j]) ? 1 : 0;
        rp[j] += cp; rp[i] += 1 - cp;
        int ct = (t[i] >= t[j]) ? 1 : 0;
        rt[j] += ct; rt[i] += 1 - ct;
      }
    }
#pragma unroll
    for (int i = 0; i < GROUP; ++i)
      ovl += ((rp[i] < TOPK) & (rt[i] < TOPK)) ? 1 : 0;
  }

  // ---- Deterministic block reduction (fixed-order) ----
  float of = (float)ovl;
  for (int off = 16; off > 0; off >>= 1) {
    mse += __shfl_xor(mse, off, 32);
    of  += __shfl_xor(of,  off, 32);
  }
  const int wave = tid >> 5;
  if ((tid & 31) == 0) { red[wave] = mse; red[8 + wave] = of; }
  __syncthreads();
  if (tid == 0) {
    float ms = 0.0f, os = 0.0f;
    for (int w = 0; w < 8; ++w) { ms += red[w]; os += red[8 + w]; }
    partials[2 * blockIdx.x]     = ms;
    partials[2 * blockIdx.x + 1] = os;
  }
}

// ---------------------------------------------------------------------------
// Kernel 2: single wave (EXEC all-ones, as WMMA requires). Grid-stride
// accumulate the per-block partials, then reduce across the wave with
// V_WMMA_F32_16X16X4_F32: with A = ones(16x4), every D entry is a column sum
// of B, so  sum(all 256 D entries) = 16 * sum(all 64 B slots)  — true for any
// operand layout. A short shuffle tree collects the lane sums.
// ---------------------------------------------------------------------------
__global__ __launch_bounds__(32)
void overlap_final(const float* __restrict__ partials,
                   float* __restrict__ out,
                   int nPart, float invN, float invMK) {
  const int lane = threadIdx.x;  // 0..31, all active
  float mse = 0.0f, ov = 0.0f;
  for (int i = lane; i < nPart; i += 32) {
    mse += partials[2 * i];
    ov  += partials[2 * i + 1];
  }
  v2f a;  a[0] = 1.0f; a[1] = 1.0f;   // A = all ones (16x4)
  v2f bm; bm[0] = mse; bm[1] = 0.0f;  // B slots carry per-lane partials
  v2f bo; bo[0] = ov;  bo[1] = 0.0f;
  v8f c = {};
  v8f dm = __builtin_amdgcn_wmma_f32_16x16x4_f32(
      false, a, false, bm, (short)0, c, false, false);
  v8f dv = __builtin_amdgcn_wmma_f32_16x16x4_f32(
      false, a, false, bo, (short)0, c, false, false);
  float sm = 0.0f, so = 0.0f;
#pragma unroll
  for (int k = 0; k < 8; ++k) { sm += dm[k]; so += dv[k]; }
  for (int off = 16; off > 0; off >>= 1) {
    sm += __shfl_xor(sm, off, 32);
    so += __shfl_xor(so, off, 32);
  }
  if (lane == 0) {
    float mseT = sm * (1.0f / 16.0f);   // undo the x16 from summing all of D
    float ovT  = so * (1.0f / 16.0f);
    out[0] = 0.6f * (mseT * invN) + 0.4f * (1.0f - ovT * invMK);
  }
}

// ---------------------------------------------------------------------------
extern "C" void kernel_launch(void* const* d_in, const int* in_sizes, int n_in,
                              void* d_out, int out_size, void* d_ws, size_t ws_size,
                              hipStream_t stream) {
  const float* pred = (const float*)d_in[0];
  const float* targ = (const float*)d_in[1];
  // d_in[2] = batch_ids (int64): groups are contiguous equal-size -> unused.
  const long long N = (long long)in_sizes[0];
  const int M = (int)(N / GROUP);
  const int nBlocks = (M + GPB - 1) / GPB;   // 1954 for M = 500k

  float* ws = (float*)d_ws;                  // 2 floats per block of partials
  overlap_main<<<nBlocks, 256, 0, stream>>>(pred, targ, ws, M, N);
  overlap_final<<<1, 32, 0, stream>>>(
      ws, (float*)d_out, nBlocks,
      (float)(1.0 / (double)N),
      (float)(1.0 / ((double)TOPK * (double)M)));
}